// VectorQuantizerEMA_35570919145946
// MI455X (gfx1250) — compile-verified
//
#include <hip/hip_runtime.h>
#include <hip/hip_bf16.h>
#include <math.h>

typedef __attribute__((ext_vector_type(2))) float v2f;
typedef __attribute__((ext_vector_type(8))) float v8f;
typedef __attribute__((ext_vector_type(4))) unsigned int v4u;
typedef __attribute__((ext_vector_type(8))) int v8i;
typedef __attribute__((ext_vector_type(4))) int v4i;

#define NB      32
#define HW      4096
#define DIM     64
#define NROWS   131072          /* 32 * 64 * 64 */
#define NCODE   256
#define DECAY_F 0.99f
#define COMMIT_F 0.25f
#define EPS_F   1e-05f

/* d_out flat offsets (reference tuple order) */
#define OFF_LOSS   0
#define OFF_OUT    1
#define OFF_PERP   8388609
#define OFF_IDX    8388610
#define OFF_NEMB   8519682
#define OFF_NCS    8536066
#define OFF_NEMA   8536322

/* dynamic LDS partition (floats). emb_sm MUST stay at offset 0: the TDM
   descriptor's lds_addr targets the dynamic-LDS base. */
#define SM_EMB    0          /* 16384 : codebook (TDM destination) */
#define SM_DW     16384      /* 16384 : segment-sum accumulator */
#define SM_CNT    32768      /* 256   : counts */
#define SM_ENORM  33024      /* 256   : ||e||^2 */
#define SM_RED    33280      /* 2048  : argmin partial values (256 thr x 8) */
#define SM_REDI   35328      /* 2048  : argmin partial indices */
#define SM_LOSS   37376      /* 1     : loss accumulator */
#define SM_TOTAL  37377

__global__ void vq_zero(float* counts, float* dw, float* loss_sum) {
    int i = blockIdx.x * 256 + threadIdx.x;
    if (i < NCODE * DIM) dw[i] = 0.0f;
    if (i < NCODE)       counts[i] = 0.0f;
    if (i == 0)          loss_sum[0] = 0.0f;
}

__global__ __launch_bounds__(256) void vq_main(
    const float* __restrict__ x,      /* NCHW [32,64,64,64] */
    const float* __restrict__ emb,    /* [256,64] */
    int*   __restrict__ idx_out,
    float* __restrict__ counts,
    float* __restrict__ dw,
    float* __restrict__ loss_sum)
{
    extern __shared__ float smem[];
    float* emb_sm   = smem + SM_EMB;
    float* dw_sm    = smem + SM_DW;
    float* cnt_sm   = smem + SM_CNT;
    float* enorm_sm = smem + SM_ENORM;
    float* redv     = smem + SM_RED;
    int*   redi     = (int*)(smem + SM_REDI);
    float* loss_sm  = smem + SM_LOSS;

    const int tid  = threadIdx.x;
    const int wave = tid >> 5;
    const int lane = tid & 31;

    /* --- Stage codebook into LDS via the Tensor Data Mover (one DMA per block).
       2D tensor: dim0=64 elems/row, dim1=256 rows, 4-byte elements; tile = whole
       tensor; destination = LDS offset 0 (emb_sm). Issued by wave 0 only (TDM
       ignores EXEC; guard prevents 8 duplicate issues). Tracked by TENSORcnt. */
    if (tid < 32) {
        unsigned long long ga = (unsigned long long)(uintptr_t)emb;
        v4u g0;
        g0.x = 1u;                                   /* count=1, is_restore=0, gather off */
        g0.y = 0u;                                   /* lds_addr = 0 (emb_sm) */
        g0.z = (unsigned)(ga & 0xFFFFFFFFu);         /* global_addr[31:0] */
        g0.w = (unsigned)((ga >> 32) & 0x01FFFFFFu)  /* global_addr[56:32] */
             | (2u << 30);                           /* type = 2 ("image") */
        v8i g1;
        g1[0] = (int)(2u << 16);                     /* wg_mask=0, data_size=2 (4B) */
        g1[1] = (int)(64u << 16);                    /* tensor_dim0 = 64 (low half) */
        g1[2] = (int)(256u << 16);                   /* dim0 hi=0 | tensor_dim1 = 256 (low half) */
        g1[3] = (int)(64u << 16);                    /* dim1 hi=0 | tile_dim0 = 64 */
        g1[4] = 256;                                 /* tile_dim1 = 256, tile_dim2 = 0 */
        g1[5] = 64;                                  /* tensor_dim0_stride = 64 */
        g1[6] = 0;
        g1[7] = 0;
        v4i gz4 = {0, 0, 0, 0};
        v8i gz8 = {0, 0, 0, 0, 0, 0, 0, 0};
        __builtin_amdgcn_tensor_load_to_lds(g0, g1, gz4, gz4, gz8, 0);
        __builtin_amdgcn_s_wait_tensorcnt(0);
    }
    /* zero block-local accumulators in parallel with nothing else touching them */
    for (int i = tid; i < NCODE * DIM; i += 256) dw_sm[i] = 0.0f;
    if (tid < NCODE) cnt_sm[tid] = 0.0f;
    if (tid == 0)    loss_sm[0] = 0.0f;
    __syncthreads();
    if (tid < NCODE) {
        float s = 0.0f;
        for (int d = 0; d < DIM; ++d) { float e = emb_sm[tid * DIM + d]; s += e * e; }
        enorm_sm[tid] = s;
    }
    __syncthreads();

    const int r     = lane & 15;
    const int khalf = (lane < 16) ? 0 : 2;

    for (int chunk = blockIdx.x; chunk < NROWS / 128; chunk += gridDim.x) {
        const int row0 = chunk * 128 + wave * 16;   /* this wave's 16 rows */
        const int n    = row0 + r;
        const int b    = n >> 12;                   /* /HW */
        const int hw   = n & 4095;
        const float* xbase = x + ((size_t)b * DIM) * HW + hw;  /* channel k at xbase[k*HW] */

        /* A fragments: 16 K-chunks of 4 (fp32 16x4 layout: lanes 0-15 K0/K1, 16-31 K2/K3) */
        v2f afrag[16];
        #pragma unroll
        for (int kc = 0; kc < 16; ++kc) {
            int k0 = kc * 4 + khalf;
            afrag[kc].x = xbase[(size_t)k0 * HW];
            afrag[kc].y = xbase[(size_t)(k0 + 1) * HW];
        }

        /* prefetch next grid-stride chunk's input lines behind the WMMA chain */
        {
            int nchunk = chunk + (int)gridDim.x;
            if (nchunk < NROWS / 128) {
                int nrow0 = nchunk * 128 + wave * 16;
                int nb2 = nrow0 >> 12, nhw = nrow0 & 4095;
                const float* nxb = x + ((size_t)nb2 * DIM) * HW + nhw;
                __builtin_prefetch(nxb + (size_t)(lane * 2) * HW, 0, 0);
                __builtin_prefetch(nxb + (size_t)(lane * 2 + 1) * HW, 0, 0);
            }
        }

        float best[8]; int besti[8];
        #pragma unroll
        for (int j = 0; j < 8; ++j) { best[j] = 3.4e38f; besti[j] = 0; }

        /* 16 code-tiles x 16 K-steps of V_WMMA_F32_16X16X4_F32 */
        for (int ct = 0; ct < 16; ++ct) {
            v8f acc = {0.f,0.f,0.f,0.f,0.f,0.f,0.f,0.f};
            const float* bbase = emb_sm + (ct * 16 + r) * DIM + khalf;
            #pragma unroll
            for (int kc = 0; kc < 16; ++kc) {
                v2f bfrag;
                bfrag.x = bbase[kc * 4 + 0];
                bfrag.y = bbase[kc * 4 + 1];
                acc = __builtin_amdgcn_wmma_f32_16x16x4_f32(
                        false, afrag[kc], false, bfrag, (short)0, acc, false, false);
            }
            const int   code = ct * 16 + r;
            const float en   = enorm_sm[code];
            #pragma unroll
            for (int j = 0; j < 8; ++j) {         /* score = ||e||^2 - 2 x.e  (||x||^2 const per row) */
                float score = en - 2.0f * acc[j];
                if (score < best[j]) { best[j] = score; besti[j] = code; }
            }
        }

        /* stage per-lane partial argmins; reduce across the 16 lanes holding each row */
        #pragma unroll
        for (int j = 0; j < 8; ++j) { redv[tid * 8 + j] = best[j]; redi[tid * 8 + j] = besti[j]; }
        __syncthreads();

        if (lane < 16) {
            const int base = wave * 32;
            const int slot = lane & 7;
            const int loff = (lane < 8) ? 0 : 16;   /* rows 8-15 live in producer lanes 16-31 */
            float bv = 3.4e38f; int bi = 0;
            for (int t = 0; t < 16; ++t) {
                float v = redv[(base + loff + t) * 8 + slot];
                int   c = redi[(base + loff + t) * 8 + slot];
                if (v < bv) { bv = v; bi = c; }
            }
            idx_out[n] = bi;
            atomicAdd(&cnt_sm[bi], 1.0f);
            float xn = 0.0f;
            for (int k = 0; k < DIM; ++k) {
                float xv = xbase[(size_t)k * HW];
                xn += xv * xv;
                atomicAdd(&dw_sm[bi * DIM + k], xv);   /* segment_sum(flat, idx) */
            }
            atomicAdd(loss_sm, xn + bv);               /* full ||x - e||^2 for the loss */
        }
        __syncthreads();
    }

    /* flush block-local accumulators */
    for (int i = tid; i < NCODE * DIM; i += 256) atomicAdd(&dw[i], dw_sm[i]);
    if (tid < NCODE) atomicAdd(&counts[tid], cnt_sm[tid]);
    if (tid == 0)    atomicAdd(loss_sum, loss_sm[0]);
}

__global__ __launch_bounds__(256) void vq_finalize(
    const float* __restrict__ counts, const float* __restrict__ dw,
    const float* __restrict__ ema_cs, const float* __restrict__ ema_w,
    const float* __restrict__ loss_sum, float* __restrict__ out)
{
    __shared__ float red[256];
    __shared__ float s_n, s_perp;
    const int t = threadIdx.x;

    float c   = counts[t];
    float raw = ema_cs[t] * DECAY_F + (1.0f - DECAY_F) * c;

    red[t] = raw; __syncthreads();
    for (int s = 128; s > 0; s >>= 1) { if (t < s) red[t] += red[t + s]; __syncthreads(); }
    if (t == 0) s_n = red[0];
    __syncthreads();

    float ncs = (raw + EPS_F) / (s_n + NCODE * EPS_F) * s_n;

    float avg = c / (float)NROWS;
    red[t] = avg * logf(avg + 1e-10f);
    __syncthreads();
    for (int s = 128; s > 0; s >>= 1) { if (t < s) red[t] += red[t + s]; __syncthreads(); }
    if (t == 0) s_perp = expf(-red[0]);
    __syncthreads();

    if (t == 0) {
        out[OFF_LOSS] = COMMIT_F * loss_sum[0] / (float)((size_t)NROWS * DIM);
        out[OFF_PERP] = s_perp;
    }
    out[OFF_NCS + t] = ncs;
    for (int d = 0; d < DIM; ++d) {
        float nw = ema_w[t * DIM + d] * DECAY_F + (1.0f - DECAY_F) * dw[t * DIM + d];
        out[OFF_NEMA + t * DIM + d] = nw;
        out[OFF_NEMB + t * DIM + d] = nw / ncs;
    }
}

__global__ void vq_out(const float* __restrict__ emb, const int* __restrict__ idx_i,
                       float* __restrict__ out)
{
    int i = blockIdx.x * 256 + threadIdx.x;
    if (i < NB * DIM * HW) {
        int b  = i >> 18;           /* / (DIM*HW) */
        int d  = (i >> 12) & 63;
        int hw = i & 4095;
        int n  = (b << 12) | hw;
        out[OFF_OUT + i] = emb[idx_i[n] * DIM + d];   /* straight-through == quantized */
    }
    if (i < NROWS) out[OFF_IDX + i] = (float)idx_i[i];
}

extern "C" void kernel_launch(void* const* d_in, const int* in_sizes, int n_in,
                              void* d_out, int out_size, void* d_ws, size_t ws_size,
                              hipStream_t stream) {
    const float* x      = (const float*)d_in[0];
    const float* emb    = (const float*)d_in[1];
    const float* ema_cs = (const float*)d_in[2];
    const float* ema_w  = (const float*)d_in[3];
    float* out = (float*)d_out;

    char* ws = (char*)d_ws;
    int*   idx_i    = (int*)ws;                                    /* 131072 ints  */
    float* counts   = (float*)(ws + (size_t)NROWS * 4);            /* 256 floats   */
    float* dw       = counts + NCODE;                              /* 16384 floats */
    float* loss_sum = dw + NCODE * DIM;                            /* 1 float      */

    vq_zero<<<(NCODE * DIM + 255) / 256, 256, 0, stream>>>(counts, dw, loss_sum);
    vq_main<<<256, 256, SM_TOTAL * sizeof(float), stream>>>(x, emb, idx_i, counts, dw, loss_sum);
    vq_finalize<<<1, 256, 0, stream>>>(counts, dw, ema_cs, ema_w, loss_sum, out);
    vq_out<<<(NB * DIM * HW + 255) / 256, 256, 0, stream>>>(emb, idx_i, out);
}